// DiscreteComms_14388140442092
// MI455X (gfx1250) — compile-verified
//
#include <hip/hip_runtime.h>

// ---- problem constants ----
#define Bsz        4096
#define Nn         8
#define Hh         512
#define Dd         512            // NUM_COMMS*COMM_SIZE
#define N_COMMS    8
#define C_SIZE     64
#define VOCAB      512
#define MROWS      (Bsz * Nn)     // 32768 logits rows
#define MB         32             // logits rows per block
#define NBLOCKS    (MROWS / MB)   // 1024
#define CBSTR      68             // padded LDS row stride (floats) -> conflict-free

// LDS layout (float offsets), dynamic shared memory
#define CB_OFF     0
#define CSQ_OFF    (CB_OFF + VOCAB * CBSTR)        // 34816
#define XS_OFF     (CSQ_OFF + VOCAB)               // 35328
#define FS_OFF     (XS_OFF + MB * CBSTR)           // 37504
#define IDX_OFF    (FS_OFF + 8 * MB * CBSTR)       // 54912
#define WL_OFF     (IDX_OFF + 8 * MB)              // 55168
#define SMEM_FLT   (WL_OFF + 8)                    // 55176 floats = 220704 B (<320KB WGP LDS)

// d_out layout: comm_output | vq_loss | comm_log_probs
#define OUT_LOSS   (MROWS * Dd)                    // 16777216
#define OUT_LOGP   (OUT_LOSS + 1)
#define LOGP_N     (Bsz * Nn)

typedef float v2f __attribute__((ext_vector_type(2)));
typedef float v8f __attribute__((ext_vector_type(8)));

__device__ __forceinline__ v8f wmma_f32(v2f a, v2f b, v8f c) {
  // D(16x16,f32) = A(16x4,f32) x B(4x16,f32) + C
  return __builtin_amdgcn_wmma_f32_16x16x4_f32(
      /*neg_a=*/false, a, /*neg_b=*/false, b,
      /*c_mod=*/(short)0, c, /*reuse_a=*/false, /*reuse_b=*/false);
}

__global__ __launch_bounds__(256) void vq_fused_kernel(
    const float* __restrict__ x, const float* __restrict__ W,
    const float* __restrict__ bias, const float* __restrict__ codebook,
    float* __restrict__ out, float* __restrict__ partials) {
  extern __shared__ float sm[];
  const int t    = threadIdx.x;
  const int wv   = t >> 5;        // wave id 0..7 == comm id
  const int lane = t & 31;
  const int r    = lane & 15;     // M (A) / N (B,C) index within tile
  const int hi   = lane >> 4;     // selects K pair {0,1} vs {2,3}
  const int kq   = hi * 2;
  const int m0   = blockIdx.x * MB;
  const int n0   = wv * C_SIZE;

  float* cb   = sm + CB_OFF;                       // codebook, padded [512][68]
  float* csq  = sm + CSQ_OFF;                      // 0.5*|c|^2 per code
  float* xs   = sm + XS_OFF;                       // x stage, padded [32][68]
  float* fs   = sm + FS_OFF + wv * (MB * CBSTR);   // this wave's flat vectors [32][68]
  int*   idxb = (int*)(sm + IDX_OFF) + wv * MB;    // per-row best code
  float* wl   = sm + WL_OFF;                       // per-wave loss partials

  // ---- Phase A: stage codebook (512x64) into padded LDS ----
  for (int i = t; i < VOCAB * 16; i += 256) {
    const int row = i >> 4, c4 = (i & 15) * 4;
    const float4 v = *(const float4*)(codebook + row * C_SIZE + c4);
    *(float4*)(cb + row * CBSTR + c4) = v;
  }
  __syncthreads();
  // half squared norms of codes
  for (int code = t; code < VOCAB; code += 256) {
    const float* cr = cb + code * CBSTR;
    float s = 0.f;
    #pragma unroll 8
    for (int k = 0; k < C_SIZE; ++k) s += cr[k] * cr[k];
    csq[code] = 0.5f * s;
  }

  // ---- Phase B: projection logits[m0:m0+32, n0:n0+64] = x @ W (WMMA f32) ----
  v8f acc[2][4];
  #pragma unroll
  for (int mi = 0; mi < 2; ++mi)
    #pragma unroll
    for (int ni = 0; ni < 4; ++ni) acc[mi][ni] = (v8f)0.f;

  for (int kc = 0; kc < Hh; kc += 64) {
    __syncthreads();  // protect xs from previous iteration's readers (covers csq at kc=0)
    for (int i = t; i < MB * 16; i += 256) {   // 32x64 chunk as float4, coalesced
      const int row = i >> 4, c4 = (i & 15) * 4;
      const float4 v = *(const float4*)(x + (size_t)(m0 + row) * Hh + kc + c4);
      *(float4*)(xs + row * CBSTR + c4) = v;
    }
    __syncthreads();
    for (int kk = 0; kk < 64; kk += 4) {
      v2f a[2];
      #pragma unroll
      for (int mi = 0; mi < 2; ++mi) {
        const float* p = xs + (mi * 16 + r) * CBSTR + kk + kq;
        a[mi] = v2f{p[0], p[1]};                 // ds_load_b64, conflict-free via pad
      }
      #pragma unroll
      for (int ni = 0; ni < 4; ++ni) {
        const float* wp = W + (size_t)(kc + kk + kq) * Dd + n0 + ni * 16 + r;
        v2f bf = v2f{wp[0], wp[Dd]};             // B[k][n], B[k+1][n]
        #pragma unroll
        for (int mi = 0; mi < 2; ++mi) acc[mi][ni] = wmma_f32(a[mi], bf, acc[mi][ni]);
      }
    }
  }

  // ---- Phase C: add bias, spill flat vectors to padded LDS ----
  float bv[4];
  #pragma unroll
  for (int ni = 0; ni < 4; ++ni) bv[ni] = bias[n0 + ni * 16 + r];
  #pragma unroll
  for (int mi = 0; mi < 2; ++mi)
    #pragma unroll
    for (int ni = 0; ni < 4; ++ni)
      #pragma unroll
      for (int vr = 0; vr < 8; ++vr) {
        const int row = mi * 16 + hi * 8 + vr;   // C/D layout: M = vr + hi*8, N = r
        fs[row * CBSTR + ni * 16 + r] = acc[mi][ni][vr] + bv[ni];
      }
  __syncthreads();

  // ---- Phase D: VQ scoring (flat @ codebook^T via WMMA) + running argmax ----
  float bestS[2][8];
  int   bestI[2][8];
  #pragma unroll
  for (int mi = 0; mi < 2; ++mi)
    #pragma unroll
    for (int vr = 0; vr < 8; ++vr) { bestS[mi][vr] = -3.4e38f; bestI[mi][vr] = VOCAB; }

  for (int g = 0; g < VOCAB / 16; ++g) {         // 32 groups of 16 codes
    v8f s[2] = {(v8f)0.f, (v8f)0.f};
    const int code = g * 16 + r;                 // this lane's candidate column
    for (int kk = 0; kk < C_SIZE; kk += 4) {
      v2f a[2];
      #pragma unroll
      for (int mi = 0; mi < 2; ++mi) {
        const float* p = fs + (mi * 16 + r) * CBSTR + kk + kq;
        a[mi] = v2f{p[0], p[1]};
      }
      const float* cp = cb + code * CBSTR + kk + kq;
      v2f bf = v2f{cp[0], cp[1]};
      #pragma unroll
      for (int mi = 0; mi < 2; ++mi) s[mi] = wmma_f32(a[mi], bf, s[mi]);
    }
    const float cq = csq[code];
    #pragma unroll
    for (int mi = 0; mi < 2; ++mi)
      #pragma unroll
      for (int vr = 0; vr < 8; ++vr) {
        const float sc = s[mi][vr] - cq;         // argmax(f·c - 0.5|c|^2) == argmin dist
        if (sc > bestS[mi][vr] ||
            (sc == bestS[mi][vr] && code < bestI[mi][vr])) {
          bestS[mi][vr] = sc; bestI[mi][vr] = code;
        }
      }
  }

  // ---- Phase E: cross-lane argmax over the 16 columns held per half-wave ----
  #pragma unroll
  for (int mi = 0; mi < 2; ++mi)
    #pragma unroll
    for (int vr = 0; vr < 8; ++vr) {
      float s = bestS[mi][vr];
      int   i = bestI[mi][vr];
      #pragma unroll
      for (int m = 1; m <= 8; m <<= 1) {         // stays within 16-lane halves
        const float so = __shfl_xor(s, m, 32);
        const int   io = __shfl_xor(i, m, 32);
        if (so > s || (so == s && io < i)) { s = so; i = io; }
      }
      if (r == 0) idxb[mi * 16 + hi * 8 + vr] = i;
    }
  __syncthreads();

  // ---- Phase F: gather codes, write straight-through output, accumulate loss ----
  float lossAcc = 0.f;
  for (int row = 0; row < MB; ++row) {
    const int idx = idxb[row];
    const float* qp = cb + idx * CBSTR;
    const float* fp = fs + row * CBSTR;
    const size_t ob = ((size_t)(m0 + row) * N_COMMS + wv) * C_SIZE;
    #pragma unroll
    for (int half = 0; half < 2; ++half) {
      const int k = lane + 32 * half;
      const float q = qp[k], f = fp[k];
      const float d = q - f;
      lossAcc += d * d;
      out[ob + k] = f + d;                       // matches reference's flat + sg(q-flat)
    }
  }
  #pragma unroll
  for (int m = 1; m <= 16; m <<= 1) lossAcc += __shfl_xor(lossAcc, m, 32);
  if (lane == 0) wl[wv] = lossAcc;
  __syncthreads();
  if (t == 0) {
    float s = 0.f;
    #pragma unroll
    for (int i = 0; i < 8; ++i) s += wl[i];
    partials[blockIdx.x] = s;                    // deterministic per-block partial
  }
}

__global__ __launch_bounds__(256) void vq_finalize_kernel(
    const float* __restrict__ partials, float* __restrict__ out) {
  __shared__ float red[256];
  const int t = threadIdx.x;
  for (int i = t; i < LOGP_N; i += 256) out[OUT_LOGP + i] = 0.f;  // comm_log_probs = 0
  float s = 0.f;
  for (int i = t; i < NBLOCKS; i += 256) s += partials[i];        // fixed order
  red[t] = s;
  __syncthreads();
  for (int st = 128; st > 0; st >>= 1) {
    if (t < st) red[t] += red[t + st];
    __syncthreads();
  }
  // vq_loss = (1 + BETA) * mean((q-f)^2) over 262144*64 elements
  if (t == 0) out[OUT_LOSS] = red[0] * (1.25f / 16777216.0f);
}

extern "C" void kernel_launch(void* const* d_in, const int* in_sizes, int n_in,
                              void* d_out, int out_size, void* d_ws, size_t ws_size,
                              hipStream_t stream) {
  (void)in_sizes; (void)n_in; (void)out_size; (void)ws_size;
  const float* x        = (const float*)d_in[0];  // (4096,1,8,512)
  const float* W        = (const float*)d_in[1];  // (512,512)
  const float* bias     = (const float*)d_in[2];  // (512,)
  const float* codebook = (const float*)d_in[3];  // (512,64)
  float* out      = (float*)d_out;
  float* partials = (float*)d_ws;                 // 1024 floats of scratch

  const size_t smem = SMEM_FLT * sizeof(float);   // ~220 KB dynamic LDS (WGP has 320 KB)
  vq_fused_kernel<<<NBLOCKS, 256, smem, stream>>>(x, W, bias, codebook, out, partials);
  vq_finalize_kernel<<<1, 256, 0, stream>>>(partials, out);
}